// GATStage6_ReduceSum_ONNX_51994874085823
// MI455X (gfx1250) — compile-verified
//
#include <hip/hip_runtime.h>

// Segment-sum (scatter-add) for GAT reduce: out[dst[e], :] += msg[e, :]
// Memory-bound: ~343 MB moved -> ~15us floor at 23.3 TB/s HBM.
// Strategy: stream msg tiles HBM -> LDS via CDNA5 async-tensor path
// (global_load_async_to_lds_b128, double-buffered with s_wait_asynccnt),
// then scatter with L2-resident global_atomic_add_f32 (output = 12.8 MB << 192 MB L2).

#define TPB 256
#define FEAT 64
#define NBLOCKS 2048
#define TILE_EDGES 80                       // 1,250,000 = 15,625 * 80 (no tail)
#define TILE_FLOATS (TILE_EDGES * FEAT)     // 5120 floats = 20 KB per buffer
#define CHUNK_FLOATS (TPB * 4)              // 1024 floats per async step (b128/thread)
#define NCHUNK (TILE_FLOATS / CHUNK_FLOATS) // 5 async issues per tile per thread

typedef int v4i __attribute__((vector_size(16)));
typedef __attribute__((address_space(1))) v4i GV4;  // global int4
typedef __attribute__((address_space(3))) v4i LV4;  // LDS int4

__device__ __forceinline__ void atomic_add_f32(float* p, float v) {
  // relaxed, agent scope, result unused -> global_atomic_add_f32 (no return)
  __hip_atomic_fetch_add(p, v, __ATOMIC_RELAXED, __HIP_MEMORY_SCOPE_AGENT);
}

__global__ __launch_bounds__(TPB) void segsum_scatter(
    const float* __restrict__ msg, const int* __restrict__ dst,
    float* __restrict__ out, int num_edges, int per_blk) {
  __shared__ float tile[2][TILE_FLOATS];  // 40 KB, double buffered
  const int tid = threadIdx.x;
  const int ntiles = num_edges / TILE_EDGES;
  const int t0 = blockIdx.x * per_blk;          // per_blk precomputed on host
  const int t1 = min(t0 + per_blk, ntiles);

  auto issue = [&](int t, int buf) {
    const float* g = msg + (size_t)t * TILE_FLOATS + tid * 4;
    float* l = &tile[buf][tid * 4];
#pragma unroll
    for (int k = 0; k < NCHUNK; ++k) {
      __builtin_amdgcn_global_load_async_to_lds_b128(
          (GV4*)(g + k * CHUNK_FLOATS),
          (LV4*)(l + k * CHUNK_FLOATS),
          /*offset=*/0, /*cpol=*/0);
    }
  };

  if (t0 < t1) issue(t0, 0);

  for (int t = t0; t < t1; ++t) {
    const int buf = (t - t0) & 1;
    if (t + 1 < t1) {
      issue(t + 1, buf ^ 1);                           // prefetch next tile
      asm volatile("s_wait_asynccnt 5" ::: "memory");  // oldest 5 (tile t) done
    } else {
      asm volatile("s_wait_asynccnt 0" ::: "memory");
    }
    __syncthreads();  // all waves' async writes for tile t visible

    const size_t e0 = (size_t)t * TILE_EDGES;
#pragma unroll
    for (int j = 0; j < NCHUNK; ++j) {
      const int task = j * TPB + tid;   // 0..1279: (edge, col4) pairs
      const int el = task >> 4;         // local edge 0..79
      const int c  = (task & 15) << 2;  // float column 0,4,...,60
      const int d  = dst[e0 + el];      // duplicate lane addrs coalesce
      const float4 v = *reinterpret_cast<const float4*>(&tile[buf][el * FEAT + c]);
      float* o = out + (size_t)d * FEAT + c;
      atomic_add_f32(o + 0, v.x);
      atomic_add_f32(o + 1, v.y);
      atomic_add_f32(o + 2, v.z);
      atomic_add_f32(o + 3, v.w);
    }
    __syncthreads();  // buf free before it becomes an async target again
  }

  // Tail edges (num_edges % TILE_EDGES): direct global path. Empty for E=1.25M.
  const size_t tail_first_task = (size_t)ntiles * TILE_EDGES * 16;
  const size_t total_tasks = (size_t)num_edges * 16;
  for (size_t task = tail_first_task + (size_t)blockIdx.x * TPB + tid;
       task < total_tasks; task += (size_t)gridDim.x * TPB) {
    const size_t e = task >> 4;
    const int c = (int)(task & 15) << 2;
    const int d = dst[e];
    const float4 v = *reinterpret_cast<const float4*>(msg + e * FEAT + c);
    float* o = out + (size_t)d * FEAT + c;
    atomic_add_f32(o + 0, v.x);
    atomic_add_f32(o + 1, v.y);
    atomic_add_f32(o + 2, v.z);
    atomic_add_f32(o + 3, v.w);
  }
}

__global__ void zero_f4(float4* __restrict__ p, int n4) {
  int i = blockIdx.x * blockDim.x + threadIdx.x;
  int stride = gridDim.x * blockDim.x;
  for (; i < n4; i += stride) p[i] = make_float4(0.f, 0.f, 0.f, 0.f);
}

extern "C" void kernel_launch(void* const* d_in, const int* in_sizes, int n_in,
                              void* d_out, int out_size, void* d_ws, size_t ws_size,
                              hipStream_t stream) {
  const float* msg = (const float*)d_in[0];     // [E, 64] f32
  const int* eidx  = (const int*)d_in[1];       // [2, E] int
  const int E = in_sizes[1] / 2;
  const int* dst = eidx + E;                    // row 1 = destination nodes
  float* out = (float*)d_out;                   // [N, 64] f32

  const int n4 = out_size / 4;
  zero_f4<<<(n4 + TPB - 1) / TPB, TPB, 0, stream>>>((float4*)d_out, n4);

  // 2048 blocks: 40 KB LDS each -> up to 8 per WGP, covers the full chip;
  // ~8 tiles per block keeps the async double-buffer pipeline busy.
  const int ntiles = E / TILE_EDGES;
  const int per_blk = (ntiles + NBLOCKS - 1) / NBLOCKS;
  segsum_scatter<<<NBLOCKS, TPB, 0, stream>>>(msg, dst, out, E, per_blk);
}